// GraphRecurrent_25967372272043
// MI455X (gfx1250) — compile-verified
//
#include <hip/hip_runtime.h>
#include <hip/hip_bf16.h>
#include <stdint.h>

#define H 128
#define GATES 512

typedef __attribute__((ext_vector_type(2))) float v2f;
typedef __attribute__((ext_vector_type(8))) float v8f;

// ---------------- consts layout within ws (floats) ----------------
// [0]=c0 [1]=c1 [2]=b_nn3.Wd1 [3]=b_nn3.Wd2
// u @ 64 (512) ; v @ 576 (512) ; p @ 1088 (128) ; q @ 1216 (128)
#define CON_U 64
#define CON_V 576
#define CON_P 1088
#define CON_Q 1216
#define CON_SZ 2048

// ---------------- K0: zero scratch ----------------
__global__ void k0_zero(float* __restrict__ p, size_t n) {
    size_t i = (size_t)blockIdx.x * blockDim.x + threadIdx.x;
    size_t stride = (size_t)gridDim.x * blockDim.x;
    for (; i < n; i += stride) p[i] = 0.0f;
}

// ---------------- K1: tiny precompute of collapsed constants ----------------
__global__ void k1_pre(const float* __restrict__ W_em, const float* __restrict__ b_em,
                       const float* __restrict__ W_lin1, const float* __restrict__ b_lin1,
                       const float* __restrict__ W_nn1, const float* __restrict__ b_nn1,
                       const float* __restrict__ W_ih, const float* __restrict__ b_ih,
                       const float* __restrict__ b_hh,
                       const float* __restrict__ W_lin3, const float* __restrict__ b_lin3,
                       const float* __restrict__ b_nn3, const float* __restrict__ W_dec,
                       float* __restrict__ con) {
    int t = threadIdx.x;
    if (t < GATES) {  // u[g], v[g]: rank-1 collapse of (s*W_nn1+b_nn1) @ W_ih
        float uu = 0.f, vv = 0.f;
        for (int k = 0; k < H; ++k) {
            float w = W_ih[k * GATES + t];
            uu += W_nn1[k] * w;
            vv += b_nn1[k] * w;
        }
        con[CON_U + t] = uu;
        con[CON_V + t] = vv + b_ih[t] + b_hh[t];
    }
    if (t < H) {      // p[j], q[j]: e3[e,:] = a_e*p + q
        float pp = 0.f, qq = 0.f;
        for (int h = 0; h < H; ++h) {
            float w = W_lin3[h * H + t];
            pp += W_em[h] * w;
            qq += b_em[h] * w;
        }
        con[CON_P + t] = pp;
        con[CON_Q + t] = qq + b_lin3[t];
    }
    if (t == 0) {     // e1[e] = a_e*c1 + c0
        float c1 = 0.f, c0 = 0.f;
        for (int h = 0; h < H; ++h) {
            c1 += W_em[h] * W_lin1[h];
            c0 += b_em[h] * W_lin1[h];
        }
        con[0] = c0 + b_lin1[0];
        con[1] = c1;
    }
    if (t == 1) {     // bias contribution to decode dots
        float s1 = 0.f, s2 = 0.f;
        for (int j = 0; j < H; ++j) {
            s1 += b_nn3[j] * W_dec[j];
            s2 += b_nn3[j] * W_dec[H + j];
        }
        con[2] = s1;
        con[3] = s2;
    }
}

// ---------------- K2: conv1 scalar message scatter ----------------
__global__ void k2_conv1(const float* __restrict__ ea, const int* __restrict__ src,
                         const int* __restrict__ dst, const float* __restrict__ x,
                         const float* __restrict__ con, float* __restrict__ agg1, int nEdges) {
    int e = blockIdx.x * blockDim.x + threadIdx.x;
    if (e >= nEdges) return;
    float m = x[src[e]] + ea[e] * con[1] + con[0];
    m = fmaxf(m, 0.0f);
    atomicAdd(&agg1[dst[e]], m);
}

// ---------------- K3: rank-1 LSTM step, writes hn and c outputs ----------------
__global__ void k3_lstm(const float* __restrict__ x, const float* __restrict__ agg1,
                        const float* __restrict__ con, float* __restrict__ hn_out,
                        float* __restrict__ c_out, int nNodes) {
    int idx = blockIdx.x * blockDim.x + threadIdx.x;
    if (idx >= nNodes * H) return;
    int n = idx >> 7;
    int k = idx & (H - 1);
    float s = x[n] + agg1[n];
    const float* u = con + CON_U;
    const float* v = con + CON_V;
    float gi = s * u[k] + v[k];                  // i gate
    float gg = s * u[2 * H + k] + v[2 * H + k];  // g gate
    float go = s * u[3 * H + k] + v[3 * H + k];  // o gate
    float ig = 1.0f / (1.0f + expf(-gi));
    float gt = tanhf(gg);
    float og = 1.0f / (1.0f + expf(-go));
    float c = ig * gt;          // f*c0 drops (c0 = 0)
    float h = og * tanhf(c);
    hn_out[idx] = h;
    c_out[idx] = c;
}

// ---------------- K4: conv3 message scatter, wave per edge, float4 lanes ----------------
__global__ __launch_bounds__(256) void k4_conv3(const float* __restrict__ ea,
                                                const int* __restrict__ src,
                                                const int* __restrict__ dst,
                                                const float* __restrict__ hn,
                                                const float* __restrict__ con,
                                                float* __restrict__ agg3, int nEdges) {
    int e = blockIdx.x * 8 + (threadIdx.x >> 5);
    if (e >= nEdges) return;
    int lane = threadIdx.x & 31;
    int c = lane * 4;
    float a = ea[e];
    int s = src[e], d = dst[e];
    float4 h4 = *(const float4*)(hn + (size_t)s * H + c);
    float4 p4 = *(const float4*)(con + CON_P + c);
    float4 q4 = *(const float4*)(con + CON_Q + c);
    float m0 = fmaxf(h4.x + a * p4.x + q4.x, 0.0f);
    float m1 = fmaxf(h4.y + a * p4.y + q4.y, 0.0f);
    float m2 = fmaxf(h4.z + a * p4.z + q4.z, 0.0f);
    float m3 = fmaxf(h4.w + a * p4.w + q4.w, 0.0f);
    float* base = agg3 + (size_t)d * H + c;
    atomicAdd(base + 0, m0);
    atomicAdd(base + 1, m1);
    atomicAdd(base + 2, m2);
    atomicAdd(base + 3, m3);
}

// ---------------- K5: fused  (hn+agg3) @ W_nn3  GEMM  +  decode dots ----------------
// V_WMMA_F32_16X16X4_F32, 8 waves/block, each wave owns a 16-row strip x 128 cols.
// Epilogue reduces each row of the accumulator tile against W_dec halves -> d1, d2.
// h3 is never materialized.
__global__ __launch_bounds__(256) void k5_gemm(const float* __restrict__ hn,
                                               const float* __restrict__ agg3,
                                               const float* __restrict__ Wg,
                                               const float* __restrict__ Wd,
                                               const float* __restrict__ con,
                                               float* __restrict__ d1, float* __restrict__ d2,
                                               int nNodes) {
    // LDS: W_nn3 staged as k-pairs: Wl[(k/2)*128 + n] = {W[k][n], W[k+1][n]}
    // -> one ds_load_b64 per B fragment, lands in an aligned VGPR pair.
    __shared__ float2 Wl[(H / 2) * H];  // 64 KB
    for (int i = threadIdx.x; i < (H / 2) * H; i += 256) {
        int kp = i >> 7;
        int n = i & (H - 1);
        Wl[i] = make_float2(Wg[(2 * kp) * H + n], Wg[(2 * kp + 1) * H + n]);
    }
    __syncthreads();

    int wave = threadIdx.x >> 5;
    int lane = threadIdx.x & 31;
    int row0 = blockIdx.x * 128 + wave * 16;
    if (row0 >= nNodes) return;    // wave-uniform early exit (EXEC stays all-ones)

    int mloc = lane & 15;          // M (for A) / N (for B) within tile
    int half = lane >> 4;          // 0: K 0,1 ; 1: K 2,3

    // clamp instead of masking: garbage A rows only affect garbage output rows
    int rowA = row0 + mloc;
    if (rowA >= nNodes) rowA = nNodes - 1;
    const float* hrow = hn + (size_t)rowA * H;
    const float* grow = agg3 + (size_t)rowA * H;

    // W_dec values this lane needs for the fused epilogue
    float wd1[8], wd2[8];
#pragma unroll
    for (int ct = 0; ct < 8; ++ct) {
        wd1[ct] = Wd[ct * 16 + mloc];
        wd2[ct] = Wd[H + ct * 16 + mloc];
    }

    v8f acc[8];
#pragma unroll
    for (int t = 0; t < 8; ++t) acc[t] = (v8f){};

    for (int kk = 0; kk < H; kk += 4) {
        int kA = kk + half * 2;
        float2 hv = *(const float2*)(hrow + kA);
        float2 gv = *(const float2*)(grow + kA);
        v2f a;
        a[0] = hv.x + gv.x;
        a[1] = hv.y + gv.y;
        int kp = (kk >> 1) + half;
#pragma unroll
        for (int ct = 0; ct < 8; ++ct) {
            float2 bf = Wl[kp * H + ct * 16 + mloc];
            v2f b;
            b[0] = bf.x;
            b[1] = bf.y;
            acc[ct] = __builtin_amdgcn_wmma_f32_16x16x4_f32(
                false, a, false, b, (short)0, acc[ct], false, false);
        }
    }

    // Fused decode: d1[row] = sum_col h3[row][col]*Wd[col] (+ bias const), same for d2.
    // C/D layout: VGPR r -> M = r + half*8, N = ct*16 + mloc.
    float cb1 = con[2], cb2 = con[3];
#pragma unroll
    for (int r = 0; r < 8; ++r) {
        float s1 = 0.f, s2 = 0.f;
#pragma unroll
        for (int ct = 0; ct < 8; ++ct) {
            s1 += acc[ct][r] * wd1[ct];
            s2 += acc[ct][r] * wd2[ct];
        }
        // reduce across the 16 lanes of this half-wave (xor offsets < 16)
        for (int off = 1; off < 16; off <<= 1) {
            s1 += __shfl_xor(s1, off, 32);
            s2 += __shfl_xor(s2, off, 32);
        }
        int row = row0 + half * 8 + r;
        if (mloc == 0 && row < nNodes) {
            d1[row] = s1 + cb1;
            d2[row] = s2 + cb2;
        }
    }
}

// ---------------- K7: edge decode out[e] = d1[src] + d2[dst] + b_dec ----------------
__global__ void k7_edge(const int* __restrict__ src, const int* __restrict__ dst,
                        const float* __restrict__ d1, const float* __restrict__ d2,
                        const float* __restrict__ b_dec, float* __restrict__ out, int nEdges) {
    int e = blockIdx.x * blockDim.x + threadIdx.x;
    if (e >= nEdges) return;
    out[e] = d1[src[e]] + d2[dst[e]] + b_dec[0];
}

static inline size_t alignUp64(size_t v) { return (v + 63) & ~(size_t)63; }

extern "C" void kernel_launch(void* const* d_in, const int* in_sizes, int n_in,
                              void* d_out, int out_size, void* d_ws, size_t ws_size,
                              hipStream_t stream) {
    const float* x      = (const float*)d_in[0];
    const float* ea     = (const float*)d_in[1];
    const int*   ei     = (const int*)d_in[2];
    const float* W_em   = (const float*)d_in[3];
    const float* b_em   = (const float*)d_in[4];
    const float* W_lin1 = (const float*)d_in[5];
    const float* b_lin1 = (const float*)d_in[6];
    const float* W_nn1  = (const float*)d_in[7];
    const float* b_nn1  = (const float*)d_in[8];
    const float* W_ih   = (const float*)d_in[9];
    /* W_hh (d_in[10]) unused: h0 = 0 */
    const float* b_ih   = (const float*)d_in[11];
    const float* b_hh   = (const float*)d_in[12];
    const float* W_lin3 = (const float*)d_in[13];
    const float* b_lin3 = (const float*)d_in[14];
    const float* W_nn3  = (const float*)d_in[15];
    const float* b_nn3  = (const float*)d_in[16];
    const float* W_dec  = (const float*)d_in[17];
    const float* b_dec  = (const float*)d_in[18];

    const int nNodes = in_sizes[0];   // 50000
    const int nEdges = in_sizes[1];   // 800000
    const int* src = ei;
    const int* dst = ei + nEdges;

    // workspace layout (floats)
    float* ws   = (float*)d_ws;
    float* con  = ws;                                  // CON_SZ
    float* agg1 = con + CON_SZ;                        // nNodes (padded)
    float* agg3 = agg1 + alignUp64((size_t)nNodes);    // nNodes*H
    float* d1   = agg3 + (size_t)nNodes * H;           // nNodes (padded)
    float* d2   = d1 + alignUp64((size_t)nNodes);      // nNodes

    // output layout: out[E], hn[N*H], c[N*H]
    float* out = (float*)d_out;
    float* hn  = out + nEdges;
    float* cc  = hn + (size_t)nNodes * H;

    // K0: zero agg1 + agg3 (contiguous)
    size_t zcount = alignUp64((size_t)nNodes) + (size_t)nNodes * H;
    k0_zero<<<2048, 256, 0, stream>>>(agg1, zcount);

    // K1: collapsed constants
    k1_pre<<<1, 512, 0, stream>>>(W_em, b_em, W_lin1, b_lin1, W_nn1, b_nn1,
                                  W_ih, b_ih, b_hh, W_lin3, b_lin3, b_nn3, W_dec, con);

    // K2: conv1 scalar scatter
    k2_conv1<<<(nEdges + 255) / 256, 256, 0, stream>>>(ea, src, dst, x, con, agg1, nEdges);

    // K3: rank-1 LSTM -> hn, c outputs
    k3_lstm<<<(nNodes * H + 255) / 256, 256, 0, stream>>>(x, agg1, con, hn, cc, nNodes);

    // K4: conv3 message scatter (wave per edge)
    k4_conv3<<<(nEdges + 7) / 8, 256, 0, stream>>>(ea, src, dst, hn, con, agg3, nEdges);

    // K5: fused WMMA GEMM + decode dots (h3 never materialized)
    k5_gemm<<<(nNodes + 127) / 128, 256, 0, stream>>>(hn, agg3, W_nn3, W_dec, con,
                                                      d1, d2, nNodes);

    // K7: edge decode
    k7_edge<<<(nEdges + 255) / 256, 256, 0, stream>>>(src, dst, d1, d2, b_dec, out, nEdges);
}